// GraphAttentionContext_1434519076918
// MI455X (gfx1250) — compile-verified
//
#include <hip/hip_runtime.h>
#include <math.h>
#include <stdint.h>

// CDNA5 fp32 WMMA: D(16x16) = A(16x4) x B(4x16) + C, wave32.
typedef __attribute__((ext_vector_type(2))) float v2f;
typedef __attribute__((ext_vector_type(8))) float v8f;

#define MSG_DIM 128
#define KN      32
#define NT      16          // nodes per workgroup tile
#define LP      132         // padded LDS row stride (keeps 16B alignment, rotates banks)
#define RSCALE  0.088388347648318447f   // 1/sqrt(128)

// ---------------------------------------------------------------------------
// Kernel 0: fold the Q/K projections into one matrix.
//   M2 = Wk^T Wq  [128x128],  b2 = Wk^T bq,  v2 = Wq^T bk,  s2 = bk.bq
// ws layout (floats): [0,16384) M2 | [16384,16512) b2 | [16512,16640) v2 | [16640] s2
// ---------------------------------------------------------------------------
__global__ __launch_bounds__(256) void gac_prep(const float* __restrict__ Wq,
                                                const float* __restrict__ bq,
                                                const float* __restrict__ Wk,
                                                const float* __restrict__ bk,
                                                float* __restrict__ ws) {
  const int idx = blockIdx.x * 256 + threadIdx.x;   // 64 blocks -> 16384 threads
  const int d = idx >> 7;
  const int e = idx & 127;
  float acc = 0.f;
  for (int a = 0; a < 128; ++a)
    acc += Wk[a * 128 + d] * Wq[a * 128 + e];
  ws[d * 128 + e] = acc;                            // M2[d][e]

  if (blockIdx.x == 0) {
    const int t = threadIdx.x;
    if (t < 128) {
      float b2 = 0.f, v2 = 0.f;
      for (int a = 0; a < 128; ++a) {
        b2 += Wk[a * 128 + t] * bq[a];
        v2 += Wq[a * 128 + t] * bk[a];
      }
      ws[16384 + t] = b2;
      ws[16512 + t] = v2;
    }
    if (t == 0) {
      float s2 = 0.f;
      for (int a = 0; a < 128; ++a) s2 += bk[a] * bq[a];
      ws[16640] = s2;
    }
  }
}

// ---------------------------------------------------------------------------
// Kernel 1: fused attention-context. One WG handles NT=16 nodes.
// X blocks stream HBM -> LDS via GLOBAL_LOAD_ASYNC_TO_LDS_B128 (ASYNCcnt).
// ---------------------------------------------------------------------------
__global__ __launch_bounds__(256) void gac_main(const float* __restrict__ q,    // [N,128]
                                                const float* __restrict__ X,    // [N,32,128]
                                                const float* __restrict__ Wv,   // [128,128]
                                                const float* __restrict__ bv,   // [128]
                                                const float* __restrict__ ws,
                                                float* __restrict__ out,        // [N,128]
                                                int N) {
  __shared__ float qs[NT][LP];        // query tile
  __shared__ float us[NT][LP];        // u = M2 q + b2 per node
  __shared__ float ys[NT][LP];        // attention-weighted neighbor sums
  __shared__ float xs[2][KN][LP];     // double-buffered neighbor block (16KB each)
  __shared__ float cn[NT];            // c = v2.q + s2 per node
  __shared__ float sc[KN];
  __shared__ float att[KN];

  const int tid  = threadIdx.x;
  const int lane = tid & 31;
  const int wave = tid >> 5;
  const int n0   = blockIdx.x * NT;

  const float* M2 = ws;
  const float* b2 = ws + 16384;
  const float* v2 = ws + 16512;
  const float  s2 = ws[16640];

  // Per-thread LDS byte offsets for the 4 async b128 transfers per buffer.
  // Generic pointer low 32 bits == workgroup-relative LDS byte address.
  uint32_t lbase[2];
  lbase[0] = (uint32_t)(uintptr_t)&xs[0][0][0];
  lbase[1] = (uint32_t)(uintptr_t)&xs[1][0][0];
  uint32_t loff[4];
  #pragma unroll
  for (int i = 0; i < 4; ++i) {
    int f = i * 1024 + tid * 4;                 // flat float index into 32x128 tile
    loff[i] = (uint32_t)(((f >> 7) * LP + (f & 127)) * 4);
  }

  // ---- load q tile (coalesced float4) ----
  for (int i = 0; i < 2; ++i) {
    int f = i * 1024 + tid * 4;                 // 2048 floats total
    int r = f >> 7, c = f & 127;
    int row = n0 + r; if (row >= N) row = N - 1;
    float4 v = *(const float4*)(q + (size_t)row * 128 + c);
    *(float4*)&qs[r][c] = v;
  }

  // ---- kick off async HBM->LDS copy of node 0's neighbor block ----
  {
    int nn = n0; if (nn >= N) nn = N - 1;
    const float* Xn = X + (size_t)nn * (KN * 128);
    #pragma unroll
    for (int i = 0; i < 4; ++i) {
      uint64_t ga = (uint64_t)(uintptr_t)(Xn + i * 1024 + tid * 4);
      uint32_t la = lbase[0] + loff[i];
      asm volatile("global_load_async_to_lds_b128 %0, %1, off"
                   :: "v"(la), "v"(ga) : "memory");
    }
  }
  __syncthreads();   // qs visible

  // ---- GEMM1 (WMMA f32): U[16x128] = qs[16x128] * M2^T + b2 ----
  // wave w owns output columns [16w, 16w+16). A row m = lane&15; lanes>=16 hold K+2.
  {
    const int m   = lane & 15;
    const int hi  = lane >> 4;
    const int col = wave * 16 + m;
    v8f acc;
    const float bini = b2[col];
    #pragma unroll
    for (int r = 0; r < 8; ++r) acc[r] = bini;
    #pragma unroll 4
    for (int kk = 0; kk < 32; ++kk) {
      const int e = kk * 4 + hi * 2;
      v2f a; a.x = qs[m][e]; a.y = qs[m][e + 1];
      const float* mrow = M2 + col * 128 + e;   // M2[col][e..e+1] (contiguous)
      v2f b; b.x = mrow[0]; b.y = mrow[1];
      acc = __builtin_amdgcn_wmma_f32_16x16x4_f32(false, a, false, b,
                                                  (short)0, acc, false, false);
    }
    #pragma unroll
    for (int r = 0; r < 8; ++r) us[r + 8 * hi][col] = acc[r];
  }

  // ---- c_n scalars (wave0 lanes 0..15) ----
  if (tid < NT) {
    float s = s2;
    for (int d = 0; d < 128; ++d) s += v2[d] * qs[tid][d];
    cn[tid] = s;
  }

  // wait for node-0 async copy, then publish everything (us, cn, xs[0])
  asm volatile("s_wait_asynccnt 0x0" ::: "memory");
  __syncthreads();

  // ---- per-node attention loop, async double-buffered X through LDS ----
  int cur = 0;
  for (int n = 0; n < NT; ++n) {
    // stream node n+1 into the back buffer while we compute on node n
    if (n + 1 < NT) {
      int nn = n0 + n + 1; if (nn >= N) nn = N - 1;
      const float* Xn = X + (size_t)nn * (KN * 128);
      const uint32_t lb = lbase[cur ^ 1];
      #pragma unroll
      for (int i = 0; i < 4; ++i) {
        uint64_t ga = (uint64_t)(uintptr_t)(Xn + i * 1024 + tid * 4);
        uint32_t la = lb + loff[i];
        asm volatile("global_load_async_to_lds_b128 %0, %1, off"
                     :: "v"(la), "v"(ga) : "memory");
      }
    }

    // scores: 8 threads per neighbor, 16 dims each, shfl-xor reduce
    {
      const int k = tid >> 3, p = tid & 7;
      const float* urow = us[n];
      const float* xrow = xs[cur][k];
      float s = 0.f;
      #pragma unroll
      for (int j = 0; j < 16; ++j) {
        int d = p * 16 + j;
        s += xrow[d] * urow[d];
      }
      s += __shfl_xor(s, 1, 32);
      s += __shfl_xor(s, 2, 32);
      s += __shfl_xor(s, 4, 32);
      if (p == 0) sc[k] = (s + cn[n]) * RSCALE;
    }
    __syncthreads();

    // softmax over 32 neighbors (wave 0, full-wave shfl reductions)
    if (wave == 0) {
      float v = sc[lane];
      float m = v;
      #pragma unroll
      for (int o = 1; o < 32; o <<= 1) m = fmaxf(m, __shfl_xor(m, o, 32));
      float e = __expf(v - m);
      float ssum = e;
      #pragma unroll
      for (int o = 1; o < 32; o <<= 1) ssum += __shfl_xor(ssum, o, 32);
      att[lane] = e / ssum;
    }
    __syncthreads();

    // y_n[d] = sum_k att[k] * X[k][d]
    if (tid < 128) {
      float y = 0.f;
      #pragma unroll
      for (int k = 0; k < KN; ++k) y += att[k] * xs[cur][k][tid];
      ys[n][tid] = y;
    }

    // ensure this wave's async writes to the back buffer have landed,
    // then publish to the whole workgroup
    if (n + 1 < NT) {
      asm volatile("s_wait_asynccnt 0x0" ::: "memory");
      cur ^= 1;
    }
    __syncthreads();
  }

  // ---- GEMM2 (WMMA f32): out[16x128] = ys[16x128] * Wv^T + bv ----
  {
    const int m   = lane & 15;
    const int hi  = lane >> 4;
    const int col = wave * 16 + m;
    v8f acc;
    const float bini = bv[col];
    #pragma unroll
    for (int r = 0; r < 8; ++r) acc[r] = bini;
    #pragma unroll 4
    for (int kk = 0; kk < 32; ++kk) {
      const int d = kk * 4 + hi * 2;
      v2f a; a.x = ys[m][d]; a.y = ys[m][d + 1];
      const float* wrow = Wv + col * 128 + d;   // Wv[col][d..d+1] (contiguous)
      v2f b; b.x = wrow[0]; b.y = wrow[1];
      acc = __builtin_amdgcn_wmma_f32_16x16x4_f32(false, a, false, b,
                                                  (short)0, acc, false, false);
    }
    #pragma unroll
    for (int r = 0; r < 8; ++r) {
      const int row = n0 + r + 8 * hi;
      if (row < N) out[(size_t)row * 128 + col] = acc[r];
    }
  }
}

// ---------------------------------------------------------------------------
extern "C" void kernel_launch(void* const* d_in, const int* in_sizes, int n_in,
                              void* d_out, int out_size, void* d_ws, size_t ws_size,
                              hipStream_t stream) {
  const float* q  = (const float*)d_in[0];   // query_msg   [N,128]
  const float* X  = (const float*)d_in[1];   // neighbor_msgs [N,32,128]
  const float* Wq = (const float*)d_in[2];
  const float* bq = (const float*)d_in[3];
  const float* Wk = (const float*)d_in[4];
  const float* bk = (const float*)d_in[5];
  const float* Wv = (const float*)d_in[6];
  const float* bv = (const float*)d_in[7];
  float* out = (float*)d_out;
  float* ws  = (float*)d_ws;   // needs 16641 floats (~65KB)

  const int N = in_sizes[0] / MSG_DIM;

  gac_prep<<<64, 256, 0, stream>>>(Wq, bq, Wk, bk, ws);
  const int nblk = (N + NT - 1) / NT;
  gac_main<<<nblk, 256, 0, stream>>>(q, X, Wv, bv, ws, out, N);
}